// GAT_49357764166009
// MI455X (gfx1250) — compile-verified
//
#include <hip/hip_runtime.h>
#include <math.h>

// ---------------------------------------------------------------------------
// GAT pipeline for MI455X (gfx1250, wave32). GEMMs use V_WMMA_F32_16X16X4_F32
// (exact fp32 — matches reference numerics) with LDS-staged tiles and fully
// unrolled K loops. Edge softmax/aggregation uses L2 atomics: the node-feature
// working set (~61 MB) fits in the 192 MB L2, so HBM (23.3 TB/s) is not the
// limiter — L2 bandwidth + atomic throughput is.
// ---------------------------------------------------------------------------

typedef __attribute__((ext_vector_type(2))) float v2f;
typedef __attribute__((ext_vector_type(8))) float v8f;

#define NNODES 30000
#define NEDGES 480000
#define NGRAPH 64
#define HFD    64
#define NHEADS 8
#define MLPD   256

#define CDIV(a, b) (((a) + (b) - 1) / (b))

__device__ __forceinline__ float warp_sum32(float v) {
#pragma unroll
  for (int o = 16; o > 0; o >>= 1) v += __shfl_xor(v, o, 32);
  return v;
}

// float atomic max via ordered int/uint trick (valid with -inf init)
__device__ __forceinline__ void atomicMaxF(float* addr, float val) {
  if (val >= 0.0f) atomicMax((int*)addr, __float_as_int(val));
  else             atomicMin((unsigned int*)addr, __float_as_uint(val));
}

// ---------------------------------------------------------------------------
// fp32 WMMA GEMM:  C[rows,cols] = act(A)[rows,K] * B[K,cols] (+bias)
// block = 128 threads (4 waves); grid = (rows/16, cols/64); wave w owns the
// 16x16 tile at columns cbase + 16*w. A tile (shared by all 4 waves) is staged
// transposed in LDS (As[k][m] -> 16 consecutive lanes hit 16 consecutive
// addresses: conflict-free). B tile staged as Bs[k][c] (contiguous 16-lane
// runs: conflict-free). Fully unrolled: K/4 v_wmma_f32_16x16x4_f32 ops.
// Fragment layouts per ISA 7.12.2:
//   A 16x4 : lane<16 -> K={kb,kb+1}, lane>=16 -> K={kb+2,kb+3}, M = lane%16
//   B 4x16 : mirrored;  C/D: VGPR r holds M = r + 8*(lane>=16), N = lane%16
// ---------------------------------------------------------------------------
template <int K, int ACT>
__global__ __launch_bounds__(128) void gemm_f32_wmma(
    const float* __restrict__ A, const float* __restrict__ B,
    const float* __restrict__ bias, float* __restrict__ C, int cols) {
  __shared__ float As[K][16];   // A transposed
  __shared__ float Bs[K][64];
  const int tid  = threadIdx.x;
  const int wave = tid >> 5;
  const int lane = tid & 31;
  const int half = lane >> 4;
  const int l16  = lane & 15;
  const int mbase = blockIdx.x * 16;
  const int cbase = blockIdx.y * 64;

  // stage A tile: 16 x K floats, float4 loads, transpose into LDS
#pragma unroll
  for (int idx = tid * 4; idx < 16 * K; idx += 128 * 4) {
    const int r = idx / K;
    const int k = idx % K;
    const float4 v = *(const float4*)(A + (size_t)(mbase + r) * K + k);
    float vv[4] = {v.x, v.y, v.z, v.w};
#pragma unroll
    for (int j = 0; j < 4; ++j) {
      float x = vv[j];
      if (ACT) x = fmaxf(x, 0.0f);
      As[k + j][r] = x;
    }
  }
  // stage B tile: K x 64 floats, float4 loads
#pragma unroll
  for (int idx = tid * 4; idx < K * 64; idx += 128 * 4) {
    const int k = idx / 64;
    const int c = idx % 64;
    *(float4*)(&Bs[k][c]) = *(const float4*)(B + (size_t)k * cols + cbase + c);
  }
  __syncthreads();

  v8f acc = {};
#pragma unroll
  for (int kb = 0; kb < K; kb += 4) {
    const int k0 = kb + 2 * half;
    v2f av = {As[k0][l16], As[k0 + 1][l16]};
    v2f bv = {Bs[k0][wave * 16 + l16], Bs[k0 + 1][wave * 16 + l16]};
    // (neg_a, A, neg_b, B, c_mod, C, reuse_a, reuse_b)
    acc = __builtin_amdgcn_wmma_f32_16x16x4_f32(false, av, false, bv, (short)0,
                                                acc, false, false);
  }

  const int col = cbase + wave * 16 + l16;
  const float bb = bias ? bias[col] : 0.0f;
#pragma unroll
  for (int r = 0; r < 8; ++r) {
    const int orow = mbase + r + 8 * half;
    C[(size_t)orow * cols + col] = acc[r] + bb;
  }
}

// ---------------------------------------------------------------------------
// small utility kernels
// ---------------------------------------------------------------------------
__global__ void fill_f32(float* __restrict__ p, float v, int n) {
  int i = blockIdx.x * 256 + threadIdx.x;
  if (i < n) p[i] = v;
}

__global__ void copy_f32(float* __restrict__ d, const float* __restrict__ s, int n) {
  int i = blockIdx.x * 256 + threadIdx.x;
  if (i < n) d[i] = s[i];
}

// segment-sum pooling: out[gid[n]*D + k] += x[n*D + k]
__global__ void pool_sum(const float* __restrict__ x, const int* __restrict__ gid,
                         float* __restrict__ out, int nodes, int D) {
  int i = blockIdx.x * 256 + threadIdx.x;
  if (i >= nodes * D) return;
  int n = i / D, k = i % D;
  atomicAdd(&out[(size_t)gid[n] * D + k], x[i]);
}

__global__ void hg_add_lrelu(float* __restrict__ hg, const float* __restrict__ t, int n) {
  int i = blockIdx.x * 256 + threadIdx.x;
  if (i < n) {
    float v = t[i];
    hg[i] += (v > 0.0f) ? v : 0.01f * v;
  }
}

// ---------------------------------------------------------------------------
// attention scores: el[n,h] = feat[n,h,:].al[h,:], er likewise
// ---------------------------------------------------------------------------
__global__ void attn_scores(const float* __restrict__ feat, const float* __restrict__ al,
                            const float* __restrict__ ar, float* __restrict__ el,
                            float* __restrict__ er, int H) {
  int i = blockIdx.x * 256 + threadIdx.x;
  if (i >= NNODES * H) return;
  int n = i / H, h = i % H;
  const float* f = feat + (size_t)n * H * HFD + h * HFD;
  const float* a = al + h * HFD;
  const float* b = ar + h * HFD;
  float s0 = 0.0f, s1 = 0.0f;
#pragma unroll 4
  for (int d = 0; d < HFD; ++d) {
    float v = f[d];
    s0 += v * a[d];
    s1 += v * b[d];
  }
  el[i] = s0;
  er[i] = s1;
}

// ---------------------------------------------------------------------------
// edge softmax passes (recompute e from el/er each pass; avoids E*H buffer)
// ---------------------------------------------------------------------------
__global__ void edge_max(const int* __restrict__ src, const int* __restrict__ dst,
                         const float* __restrict__ el, const float* __restrict__ er,
                         float* __restrict__ m, int H) {
  int i = blockIdx.x * 256 + threadIdx.x;
  if (i >= NEDGES * H) return;
  int e = i / H, h = i % H;
  int s = src[e], d = dst[e];
  float v = el[s * H + h] + er[d * H + h];
  v = (v > 0.0f) ? v : 0.2f * v;
  atomicMaxF(&m[d * H + h], v);
}

__global__ void edge_expsum(const int* __restrict__ src, const int* __restrict__ dst,
                            const float* __restrict__ el, const float* __restrict__ er,
                            const float* __restrict__ m, float* __restrict__ den, int H) {
  int i = blockIdx.x * 256 + threadIdx.x;
  if (i >= NEDGES * H) return;
  int e = i / H, h = i % H;
  int s = src[e], d = dst[e];
  float v = el[s * H + h] + er[d * H + h];
  v = (v > 0.0f) ? v : 0.2f * v;
  atomicAdd(&den[d * H + h], __expf(v - m[d * H + h]));
}

// scatter-aggregate: rst[dst,h,:] += softmax_w * feat[src,h,:]  (rst pre-seeded
// with the residual term). One thread per (edge, head, d) -> coalesced feat
// loads and coalesced L2 atomics.
__global__ void edge_aggregate(const int* __restrict__ src, const int* __restrict__ dst,
                               const float* __restrict__ el, const float* __restrict__ er,
                               const float* __restrict__ m, const float* __restrict__ den,
                               const float* __restrict__ feat, float* __restrict__ rst,
                               int H) {
  long long i = (long long)blockIdx.x * 256 + threadIdx.x;
  long long tot = (long long)NEDGES * H * HFD;
  if (i >= tot) return;
  int e  = (int)(i / (H * HFD));
  int r  = (int)(i % (H * HFD));
  int h  = r / HFD;
  int dd = r % HFD;
  int s = src[e], d = dst[e];
  float v = el[s * H + h] + er[d * H + h];
  v = (v > 0.0f) ? v : 0.2f * v;
  float w = __expf(v - m[d * H + h]) / den[d * H + h];
  atomicAdd(&rst[(size_t)d * H * HFD + h * HFD + dd],
            w * feat[(size_t)s * H * HFD + h * HFD + dd]);
}

// ---------------------------------------------------------------------------
// layer-1 epilogue: per (node, head): ELU(rst + b1) -> LayerNorm(d) -> mean(heads)
// one block per node; wave w handles head w; lane holds d={lane, lane+32}.
// ---------------------------------------------------------------------------
__global__ void gat1_epilogue(const float* __restrict__ rst, const float* __restrict__ b1,
                              float* __restrict__ h1) {
  __shared__ float acc[HFD];
  const int n    = blockIdx.x;
  const int head = threadIdx.x >> 5;
  const int lane = threadIdx.x & 31;
  if (threadIdx.x < HFD) acc[threadIdx.x] = 0.0f;
  __syncthreads();
  const size_t base = (size_t)n * NHEADS * HFD + head * HFD;
  float x0 = rst[base + lane]      + b1[head * HFD + lane];
  float x1 = rst[base + lane + 32] + b1[head * HFD + lane + 32];
  x0 = (x0 > 0.0f) ? x0 : (__expf(x0) - 1.0f);
  x1 = (x1 > 0.0f) ? x1 : (__expf(x1) - 1.0f);
  const float mu  = warp_sum32(x0 + x1) * (1.0f / HFD);
  const float d0  = x0 - mu, d1 = x1 - mu;
  const float var = warp_sum32(d0 * d0 + d1 * d1) * (1.0f / HFD);
  const float inv = rsqrtf(var + 1e-5f);
  atomicAdd(&acc[lane],      d0 * inv);
  atomicAdd(&acc[lane + 32], d1 * inv);
  __syncthreads();
  if (threadIdx.x < HFD) h1[(size_t)n * HFD + threadIdx.x] = acc[threadIdx.x] * (1.0f / NHEADS);
}

// layer-2 epilogue: 1 head -> ELU(rst2 + b2) -> LayerNorm. One wave per node.
__global__ void gat2_epilogue(const float* __restrict__ rst2, const float* __restrict__ b2,
                              float* __restrict__ h2) {
  const int wave = threadIdx.x >> 5;
  const int lane = threadIdx.x & 31;
  const int n = blockIdx.x * 8 + wave;
  if (n >= NNODES) return;
  float x0 = rst2[(size_t)n * HFD + lane]      + b2[lane];
  float x1 = rst2[(size_t)n * HFD + lane + 32] + b2[lane + 32];
  x0 = (x0 > 0.0f) ? x0 : (__expf(x0) - 1.0f);
  x1 = (x1 > 0.0f) ? x1 : (__expf(x1) - 1.0f);
  const float mu  = warp_sum32(x0 + x1) * (1.0f / HFD);
  const float d0  = x0 - mu, d1 = x1 - mu;
  const float var = warp_sum32(d0 * d0 + d1 * d1) * (1.0f / HFD);
  const float inv = rsqrtf(var + 1e-5f);
  h2[(size_t)n * HFD + lane]      = d0 * inv;
  h2[(size_t)n * HFD + lane + 32] = d1 * inv;
}

// ---------------------------------------------------------------------------
extern "C" void kernel_launch(void* const* d_in, const int* in_sizes, int n_in,
                              void* d_out, int out_size, void* d_ws, size_t ws_size,
                              hipStream_t stream) {
  (void)in_sizes; (void)n_in; (void)out_size; (void)ws_size;

  const float* X     = (const float*)d_in[0];
  const int*   src   = (const int*)d_in[1];
  const int*   dst   = (const int*)d_in[2];
  const int*   gid   = (const int*)d_in[3];
  const float* projW = (const float*)d_in[4];
  const float* projb = (const float*)d_in[5];
  const float* fcW1  = (const float*)d_in[6];
  const float* al1   = (const float*)d_in[7];
  const float* ar1   = (const float*)d_in[8];
  const float* resW1 = (const float*)d_in[9];
  const float* b1    = (const float*)d_in[10];
  const float* fcW2  = (const float*)d_in[11];
  const float* al2   = (const float*)d_in[12];
  const float* ar2   = (const float*)d_in[13];
  const float* b2    = (const float*)d_in[14];
  const float* gW1   = (const float*)d_in[15];
  const float* gb1   = (const float*)d_in[16];
  const float* gW2   = (const float*)d_in[17];
  const float* gb2   = (const float*)d_in[18];
  const float* mW0   = (const float*)d_in[19];
  const float* mb0   = (const float*)d_in[20];
  const float* mW1   = (const float*)d_in[21];
  const float* mb1   = (const float*)d_in[22];
  const float* mW2   = (const float*)d_in[23];
  const float* mb2   = (const float*)d_in[24];
  float* out = (float*)d_out;

  // ---- workspace layout (floats) ----
  float* ws = (float*)d_ws;
  size_t o = 0;
  float* h0    = ws + o; o += (size_t)NNODES * HFD;           // [N,64]
  float* feat1 = ws + o; o += (size_t)NNODES * NHEADS * HFD;  // [N,512]
  float* rst1  = ws + o; o += (size_t)NNODES * NHEADS * HFD;  // [N,512] seeded w/ residual
  float* el1   = ws + o; o += (size_t)NNODES * NHEADS;
  float* er1   = ws + o; o += (size_t)NNODES * NHEADS;
  float* m1    = ws + o; o += (size_t)NNODES * NHEADS;
  float* den1  = ws + o; o += (size_t)NNODES * NHEADS;
  float* h1    = ws + o; o += (size_t)NNODES * HFD;
  float* feat2 = ws + o; o += (size_t)NNODES * HFD;
  float* rst2  = ws + o; o += (size_t)NNODES * HFD;           // seeded w/ h1 (identity res)
  float* el2   = ws + o; o += (size_t)NNODES;
  float* er2   = ws + o; o += (size_t)NNODES;
  float* m2    = ws + o; o += (size_t)NNODES;
  float* den2  = ws + o; o += (size_t)NNODES;
  float* hg    = ws + o; o += (size_t)NGRAPH * HFD;           // [64,64]
  float* pt    = ws + o; o += (size_t)NGRAPH * HFD;           // pooled tmp
  float* pt2   = ws + o; o += (size_t)NGRAPH * HFD;           // gemm tmp
  float* t0    = ws + o; o += (size_t)NGRAPH * MLPD;          // [64,256]
  float* t1    = ws + o; o += (size_t)NGRAPH * MLPD;          // [64,256]

  const float NEG_INF = -INFINITY;

  // 1) h0 = X @ projW + projb      [30000,64]x[64,64]
  gemm_f32_wmma<64, 0><<<dim3(NNODES / 16, 1), 128, 0, stream>>>(X, projW, projb, h0, HFD);
  // 2) hg = pool(h0)
  fill_f32<<<CDIV(NGRAPH * HFD, 256), 256, 0, stream>>>(hg, 0.0f, NGRAPH * HFD);
  pool_sum<<<CDIV(NNODES * HFD, 256), 256, 0, stream>>>(h0, gid, hg, NNODES, HFD);

  // 3) GAT layer 1
  gemm_f32_wmma<64, 0><<<dim3(NNODES / 16, 8), 128, 0, stream>>>(h0, fcW1, nullptr, feat1,
                                                                 NHEADS * HFD);
  gemm_f32_wmma<64, 0><<<dim3(NNODES / 16, 8), 128, 0, stream>>>(h0, resW1, nullptr, rst1,
                                                                 NHEADS * HFD);
  attn_scores<<<CDIV(NNODES * NHEADS, 256), 256, 0, stream>>>(feat1, al1, ar1, el1, er1, NHEADS);
  fill_f32<<<CDIV(NNODES * NHEADS, 256), 256, 0, stream>>>(m1, NEG_INF, NNODES * NHEADS);
  fill_f32<<<CDIV(NNODES * NHEADS, 256), 256, 0, stream>>>(den1, 0.0f, NNODES * NHEADS);
  edge_max<<<CDIV(NEDGES * NHEADS, 256), 256, 0, stream>>>(src, dst, el1, er1, m1, NHEADS);
  edge_expsum<<<CDIV(NEDGES * NHEADS, 256), 256, 0, stream>>>(src, dst, el1, er1, m1, den1, NHEADS);
  {
    long long tot = (long long)NEDGES * NHEADS * HFD;
    edge_aggregate<<<(unsigned)((tot + 255) / 256), 256, 0, stream>>>(src, dst, el1, er1, m1,
                                                                     den1, feat1, rst1, NHEADS);
  }
  gat1_epilogue<<<NNODES, 256, 0, stream>>>(rst1, b1, h1);

  // 4) hg += leaky_relu(pool(h1) @ gW1 + gb1, 0.01)
  fill_f32<<<CDIV(NGRAPH * HFD, 256), 256, 0, stream>>>(pt, 0.0f, NGRAPH * HFD);
  pool_sum<<<CDIV(NNODES * HFD, 256), 256, 0, stream>>>(h1, gid, pt, NNODES, HFD);
  gemm_f32_wmma<64, 0><<<dim3(NGRAPH / 16, 1), 128, 0, stream>>>(pt, gW1, gb1, pt2, HFD);
  hg_add_lrelu<<<CDIV(NGRAPH * HFD, 256), 256, 0, stream>>>(hg, pt2, NGRAPH * HFD);

  // 5) GAT layer 2 (1 head, identity residual)
  gemm_f32_wmma<64, 0><<<dim3(NNODES / 16, 1), 128, 0, stream>>>(h1, fcW2, nullptr, feat2, HFD);
  attn_scores<<<CDIV(NNODES, 256), 256, 0, stream>>>(feat2, al2, ar2, el2, er2, 1);
  copy_f32<<<CDIV(NNODES * HFD, 256), 256, 0, stream>>>(rst2, h1, NNODES * HFD);
  fill_f32<<<CDIV(NNODES, 256), 256, 0, stream>>>(m2, NEG_INF, NNODES);
  fill_f32<<<CDIV(NNODES, 256), 256, 0, stream>>>(den2, 0.0f, NNODES);
  edge_max<<<CDIV(NEDGES, 256), 256, 0, stream>>>(src, dst, el2, er2, m2, 1);
  edge_expsum<<<CDIV(NEDGES, 256), 256, 0, stream>>>(src, dst, el2, er2, m2, den2, 1);
  {
    long long tot = (long long)NEDGES * HFD;
    edge_aggregate<<<(unsigned)((tot + 255) / 256), 256, 0, stream>>>(src, dst, el2, er2, m2,
                                                                     den2, feat2, rst2, 1);
  }
  gat2_epilogue<<<NNODES / 8, 256, 0, stream>>>(rst2, b2, h1 /*reuse h1 as h2*/);

  // 6) hg += leaky_relu(pool(h2) @ gW2 + gb2, 0.01)
  fill_f32<<<CDIV(NGRAPH * HFD, 256), 256, 0, stream>>>(pt, 0.0f, NGRAPH * HFD);
  pool_sum<<<CDIV(NNODES * HFD, 256), 256, 0, stream>>>(h1, gid, pt, NNODES, HFD);
  gemm_f32_wmma<64, 0><<<dim3(NGRAPH / 16, 1), 128, 0, stream>>>(pt, gW2, gb2, pt2, HFD);
  hg_add_lrelu<<<CDIV(NGRAPH * HFD, 256), 256, 0, stream>>>(hg, pt2, NGRAPH * HFD);

  // 7) MLP head: [64,64]->[64,256]->[64,256]->[64,256]
  gemm_f32_wmma<64, 0><<<dim3(NGRAPH / 16, MLPD / 64), 128, 0, stream>>>(hg, mW0, mb0, t0, MLPD);
  gemm_f32_wmma<256, 1><<<dim3(NGRAPH / 16, MLPD / 64), 128, 0, stream>>>(t0, mW1, mb1, t1, MLPD);
  gemm_f32_wmma<256, 1><<<dim3(NGRAPH / 16, MLPD / 64), 128, 0, stream>>>(t1, mW2, mb2, out, MLPD);
}